// ToothLandmarkLoss_17411797418430
// MI455X (gfx1250) — compile-verified
//
#include <hip/hip_runtime.h>

typedef __attribute__((ext_vector_type(2))) float v2f;
typedef __attribute__((ext_vector_type(8))) float v8f;

#define D_ 96
#define H_ 128
#define W_ 128
#define SLAB_ELEMS (D_ * H_ * W_)
#define SLABS 64
#define CHUNKS 24   // 24 chunks * 8 waves * 4 groups = 768 row-groups = 96*8

// ---------------------------------------------------------------------------
// Stage 1: per-slab moment reduction, matmul-structured.
// For each 16-row tile (rows = h within one d-plane):
//   D(16x16) += A(16x4) * B(4x16), B col0 = 1, col1 = x-index, cols 2..15 = 0
// => accumulator col0 holds per-row sums, col1 holds per-row x-weighted sums.
// h and d weights are constant per row -> folded after the K loop.
// B is maintained incrementally (+4 per K-step on the x-weight column) so the
// hot loop is just: global_load_b64, 2x v_add_f32, v_wmma.  Two independent
// accumulator chains (even/odd k) break the WMMA RAW dependency chain.
// ---------------------------------------------------------------------------
__global__ __launch_bounds__(256)
void tooth_stage1(const float* __restrict__ seg, float* __restrict__ part) {
  const int slab  = blockIdx.x;    // b*32 + c
  const int chunk = blockIdx.y;    // 0..CHUNKS-1
  const int tid   = threadIdx.x;
  const int wave  = tid >> 5;      // 0..7
  const int lane  = tid & 31;
  const int col   = lane & 15;     // A: M index; B/D: N index
  const int hi    = lane >> 4;     // selects K half (A/B) and M half (D)

  const float* slabp = seg + (size_t)slab * SLAB_ELEMS;

  const bool isC0 = (col == 0);
  const bool isC1 = (col == 1);

  // Incremental B fragment: col0 lanes hold (1,1) forever (step 0),
  // col1 lanes start at (2*hi, 2*hi+1) and step +4 per K-chunk,
  // all other lanes hold (0,0) (step 0).
  const float bx0  = isC0 ? 1.f : (isC1 ? (float)(2 * hi)       : 0.f);
  const float by0  = isC0 ? 1.f : (isC1 ? (float)(2 * hi) + 1.f : 0.f);
  const float bstp = isC1 ? 4.f : 0.f;

  float pA = 0.f;   // -> S0 (col0 lanes) or Sx (col1 lanes)
  float pB = 0.f;   // -> Sy (col0 lanes)
  float pC = 0.f;   // -> Sz (col0 lanes)

  for (int i = 0; i < 4; ++i) {
    const int rg    = chunk * 32 + wave * 4 + i;   // row-group 0..767
    const int d     = rg >> 3;                     // d-plane 0..95
    const int hbase = (rg & 7) << 4;               // 16-row tile base within plane
    const int row   = hbase + col;                 // A-matrix row M = col
    const v2f* rp = reinterpret_cast<const v2f*>(
        slabp + (size_t)d * (H_ * W_) + (size_t)row * W_ + hi * 2);

    v8f ce = {0.f, 0.f, 0.f, 0.f, 0.f, 0.f, 0.f, 0.f};
    v8f co = {0.f, 0.f, 0.f, 0.f, 0.f, 0.f, 0.f, 0.f};
    v2f b;
    b.x = bx0;
    b.y = by0;
#pragma unroll 4
    for (int k = 0; k < 32; k += 2) {
      const v2f a0 = rp[k * 2];
      const v2f a1 = rp[k * 2 + 2];
      ce = __builtin_amdgcn_wmma_f32_16x16x4_f32(
               false, a0, false, b, (short)0, ce, false, false);
      b.x += bstp; b.y += bstp;
      co = __builtin_amdgcn_wmma_f32_16x16x4_f32(
               false, a1, false, b, (short)0, co, false, false);
      b.x += bstp; b.y += bstp;
    }
    const v8f c = ce + co;
    // c[r] = D[r + 8*hi][col]: row sums (col==0) or x-sums (col==1); 0 otherwise
    const float sum  = c[0] + c[1] + c[2] + c[3] + c[4] + c[5] + c[6] + c[7];
    const float wsum = c[1] + 2.f*c[2] + 3.f*c[3] + 4.f*c[4]
                            + 5.f*c[5] + 6.f*c[6] + 7.f*c[7];
    const float rowoff = (float)(hbase + hi * 8);
    pA += sum;                    // S0 partial (col0) / Sx partial (col1)
    pB += rowoff * sum + wsum;    // sum_r h(r) * rowsum(r)
    pC += (float)d * sum;         // d * sum
  }

  const float v0 = isC0 ? pA : 0.f;   // S0
  const float v1 = isC1 ? pA : 0.f;   // Sx
  const float v2 = isC0 ? pB : 0.f;   // Sy
  const float v3 = isC0 ? pC : 0.f;   // Sz

  __shared__ float red[256][4];
  red[tid][0] = v0; red[tid][1] = v1; red[tid][2] = v2; red[tid][3] = v3;
  __syncthreads();
  for (int s = 128; s > 0; s >>= 1) {
    if (tid < s) {
      red[tid][0] += red[tid + s][0];
      red[tid][1] += red[tid + s][1];
      red[tid][2] += red[tid + s][2];
      red[tid][3] += red[tid + s][3];
    }
    __syncthreads();
  }
  if (tid == 0) {
    float* o = part + ((size_t)slab * CHUNKS + chunk) * 4;
    o[0] = red[0][0]; o[1] = red[0][1]; o[2] = red[0][2]; o[3] = red[0][3];
  }
}

// ---------------------------------------------------------------------------
// Stage 2: deterministic partial reduction + full loss epilogue.
// ---------------------------------------------------------------------------
__global__ __launch_bounds__(64)
void tooth_stage2(const float* __restrict__ part, float* __restrict__ out) {
  __shared__ float cx[2][32], cy[2][32], cz[2][32], vol[2][32];
  const int t = threadIdx.x;            // slab id 0..63
  float S0 = 0.f, Sx = 0.f, Sy = 0.f, Sz = 0.f;
  for (int j = 0; j < CHUNKS; ++j) {    // fixed order -> deterministic
    const float* p = part + ((size_t)t * CHUNKS + j) * 4;
    S0 += p[0]; Sx += p[1]; Sy += p[2]; Sz += p[3];
  }
  const int b = t >> 5, c = t & 31;
  const float safe = (S0 > 0.f) ? S0 : 1.f;
  const bool  ok   = (S0 > 0.f);
  cx[b][c]  = ok ? Sx / safe / (float)W_ : 0.f;
  cy[b][c]  = ok ? Sy / safe / (float)H_ : 0.f;
  cz[b][c]  = ok ? Sz / safe / (float)D_ : 0.f;
  vol[b][c] = S0;
  __syncthreads();

  if (t == 0) {
    const float PI = 3.14159265358979323846f;
    float ex[32], ey[32], ez[32];
    for (int i = 0; i < 16; ++i) {
      const float ua = -PI * 0.5f + (PI / 15.f) * (float)i;
      ex[i] = 0.6f * __cosf(ua); ey[i] = 0.6f * __sinf(ua); ez[i] = 0.7f;
      const float la = PI * 0.5f - (PI / 15.f) * (float)i;
      ex[16 + i] = 0.55f * __cosf(la); ey[16 + i] = 0.55f * __sinf(la);
      ez[16 + i] = 0.3f;
    }

    // landmark: mean over (2,32,3)
    float lm = 0.f;
    for (int bb = 0; bb < 2; ++bb)
      for (int i = 0; i < 32; ++i) {
        const float dx = cx[bb][i] - ex[i];
        const float dy = cy[bb][i] - ey[i];
        const float dz = cz[bb][i] - ez[i];
        lm += dx * dx + dy * dy + dz * dz;
      }
    lm /= 192.f;

    // adjacent pairs: (i,i+1), i=0..14 except 7, both halves; mean over batch
    float spa = 0.f;
    for (int half = 0; half < 2; ++half)
      for (int i = 0; i < 15; ++i) {
        if (i == 7) continue;
        const int a1 = half * 16 + i, a2 = a1 + 1;
        for (int bb = 0; bb < 2; ++bb) {
          const float dx = cx[bb][a1] - cx[bb][a2];
          const float dy = cy[bb][a1] - cy[bb][a2];
          const float dz = cz[bb][a1] - cz[bb][a2];
          const float r  = sqrtf(dx * dx + dy * dy + dz * dz) - 0.1f;
          spa += (r > 0.f ? r : 0.f) * 0.5f;
        }
      }

    // opposing pairs (i, 15-i) per half: xy/z losses + symmetry yz/x losses
    float xy = 0.f, zl = 0.f, yz = 0.f, xs = 0.f;
    for (int half = 0; half < 2; ++half)
      for (int i = 0; i < 8; ++i) {
        const int o1 = half * 16 + i, o2 = half * 16 + 15 - i;
        for (int bb = 0; bb < 2; ++bb) {
          const float dx = cx[bb][o1] - cx[bb][o2];
          const float dy = cy[bb][o1] - cy[bb][o2];
          const float dz = cz[bb][o1] - cz[bb][o2];
          xy += (dx * dx + dy * dy) * 0.25f;        // mean over (batch, 2 comps)
          const float r = 0.3f - fabsf(dz);
          zl += (r > 0.f ? r : 0.f) * 0.5f;         // mean over batch
          yz += (dy * dy + dz * dz) * 0.25f;
          const float sx_ = cx[bb][o1] + cx[bb][o2];
          xs += sx_ * sx_ * 0.5f;
        }
      }

    // anatomy: sample variance (ddof=1) per tooth-type group, mean over batch
    const int g1[8]  = {6, 7, 8, 9, 22, 23, 24, 25};
    const int g2[4]  = {5, 10, 21, 26};
    const int g3[8]  = {3, 4, 11, 12, 19, 20, 27, 28};
    const int g4[12] = {0, 1, 2, 13, 14, 15, 16, 17, 18, 29, 30, 31};
    auto group_var = [&](const int* idx, int n) -> float {
      float acc = 0.f;
      for (int bb = 0; bb < 2; ++bb) {
        float m = 0.f;
        for (int i = 0; i < n; ++i) m += vol[bb][idx[i]];
        m /= (float)n;
        float v = 0.f;
        for (int i = 0; i < n; ++i) {
          const float d = vol[bb][idx[i]] - m;
          v += d * d;
        }
        acc += (v / (float)(n - 1)) * 0.5f;
      }
      return acc;
    };
    float anat = group_var(g1, 8) + group_var(g2, 4)
               + group_var(g3, 8) + group_var(g4, 12);

    const float landmark = lm * 10.f;
    const float spatial  = (spa + xy + zl) * 5.f;
    const float symmetry = (yz + xs) * 3.f;
    const float anatomy  = anat * 7.f;
    out[0] = landmark;
    out[1] = spatial;
    out[2] = symmetry;
    out[3] = anatomy;
    out[4] = landmark + spatial + symmetry + anatomy;
  }
}

extern "C" void kernel_launch(void* const* d_in, const int* in_sizes, int n_in,
                              void* d_out, int out_size, void* d_ws, size_t ws_size,
                              hipStream_t stream) {
  (void)in_sizes; (void)n_in; (void)out_size; (void)ws_size;
  const float* seg = (const float*)d_in[0];
  float* part = (float*)d_ws;                 // 64 * 24 * 4 floats = 24 KB
  tooth_stage1<<<dim3(SLABS, CHUNKS), 256, 0, stream>>>(seg, part);
  tooth_stage2<<<1, 64, 0, stream>>>(part, (float*)d_out);
}